// get_global_21990232555767
// MI455X (gfx1250) — compile-verified
//
#include <hip/hip_runtime.h>
#include <hip/hip_bf16.h>

typedef __attribute__((ext_vector_type(2))) float v2f;
typedef __attribute__((ext_vector_type(8))) float v8f;

#define BNS 0.9999950000374997f   // 1/sqrt(1+1e-5)

__device__ __forceinline__ v8f wmma4f32(v2f a, v2f b, v8f c) {
  return __builtin_amdgcn_wmma_f32_16x16x4_f32(false, a, false, b, (short)0, c,
                                               false, false);
}

// ---------------------------------------------------------------------------
// Generic strided WMMA GEMM:  C[b,m,n] = epi( sum_k A[b,m,k] * B[b,n,k] )
// Each wave computes a 32(m) x 16(n) tile (two accumulators, shared B frag).
//   A elem = A[b*Abs + m*As0 + k*As1], B elem = B[b*Bbs + n*Bs0 + k*Bs1]
// mode 0: store acc
// mode 1: store acc + bias[n]
// mode 2: store resid[m,n] - rowscale[m]*acc            (x - x_r)
// mode 3: store resid[m,n] + relu((acc+bias[n])*g*BNS + b)
// mode 4: store lrelu(acc*g*BNS + b, 0.2)
// Requires M%32==0, N%16==0, K%4==0.
// ---------------------------------------------------------------------------
__global__ __launch_bounds__(32)
void wmma_gemm(const float* __restrict__ A, long Abs, long As0, long As1,
               const float* __restrict__ Bm, long Bbs, long Bs0, long Bs1,
               float* __restrict__ O, long Obs,
               int M, int N, int K, int mode,
               const float* __restrict__ bias,
               const float* __restrict__ gam,
               const float* __restrict__ bet,
               const float* __restrict__ resid, long Rbs,
               const float* __restrict__ rsc, long Sbs)
{
  const int lane = threadIdx.x;
  const int half = lane >> 4;
  const int lm   = lane & 15;
  const int ntiles = N >> 4;
  const int mt = blockIdx.x / ntiles;
  const int nt = blockIdx.x % ntiles;
  const int b  = blockIdx.y;
  const int m0 = mt << 5, n0 = nt << 4;

  const float* A0 = A + (long)b * Abs + (long)(m0 + lm) * As0;
  const float* A1 = A0 + (long)16 * As0;
  const float* Bb = Bm + (long)b * Bbs + (long)(n0 + lm) * Bs0;

  v8f acc0 = {0.f, 0.f, 0.f, 0.f, 0.f, 0.f, 0.f, 0.f};
  v8f acc1 = {0.f, 0.f, 0.f, 0.f, 0.f, 0.f, 0.f, 0.f};

  if (As1 == 1 && Bs1 == 1) {          // contiguous-K fast path (b64 loads)
    for (int kb = 0; kb < K; kb += 4) {
      const int ka = kb + 2 * half;
      const v2f a0 = *(const v2f*)(A0 + ka);
      const v2f a1 = *(const v2f*)(A1 + ka);
      const v2f bv = *(const v2f*)(Bb + ka);
      acc0 = wmma4f32(a0, bv, acc0);
      acc1 = wmma4f32(a1, bv, acc1);
    }
  } else {                             // strided path (transposed operands)
    for (int kb = 0; kb < K; kb += 4) {
      const int ka = kb + 2 * half;
      v2f a0, a1, bv;
      const float* p0 = A0 + (long)ka * As1;
      const float* p1 = A1 + (long)ka * As1;
      const float* pb = Bb + (long)ka * Bs1;
      a0.x = p0[0];  a0.y = p0[As1];
      a1.x = p1[0];  a1.y = p1[As1];
      bv.x = pb[0];  bv.y = pb[Bs1];
      acc0 = wmma4f32(a0, bv, acc0);
      acc1 = wmma4f32(a1, bv, acc1);
    }
  }

  const int nn = n0 + lm;
  float bi = 0.f, gs = 0.f, bs = 0.f;
  if (mode == 1 || mode == 3) bi = bias[nn];
  if (mode == 3 || mode == 4) { gs = gam[nn] * BNS; bs = bet[nn]; }

  float* Ob = O + (long)b * Obs;
  const float* Rb = (mode == 2 || mode == 3) ? (resid + (long)b * Rbs) : nullptr;

  auto emit = [&](const v8f& acc, int mbase) {
#pragma unroll
    for (int r = 0; r < 8; ++r) {
      const int  mm = mbase + half * 8 + r;
      const long oi = (long)mm * N + nn;
      const float v = acc[r];
      float o;
      if (mode == 0)      o = v;
      else if (mode == 1) o = v + bi;
      else if (mode == 2) o = Rb[oi] - rsc[(long)b * Sbs + mm] * v;
      else if (mode == 3) { float d = (v + bi) * gs + bs; o = Rb[oi] + (d > 0.f ? d : 0.f); }
      else                { float d = v * gs + bs;        o = (d > 0.f) ? d : 0.2f * d; }
      Ob[oi] = o;
    }
  };
  emit(acc0, m0);
  emit(acc1, m0 + 16);
}

// ---------------------------------------------------------------------------
// Gather + 1x1 conv + BN + LeakyReLU(0.2) + max over G groups (WMMA).
// Each wave computes a 16(l) x 32(o) tile: gathered A fragment reused across
// two weight fragments held in LDS.
// X elem(b,row,c) = X[b*xbs + row*xrs + c*xcs]; idx: [B, G*L]; W: [cout,cin]
// O: [B, L, cout].
// ---------------------------------------------------------------------------
__global__ __launch_bounds__(32)
void stage_gcmax(const float* __restrict__ X, long xbs, long xrs, long xcs,
                 const int* __restrict__ idx,
                 const float* __restrict__ W,
                 const float* __restrict__ gam, const float* __restrict__ bet,
                 float* __restrict__ O,
                 int L, int cin, int cout, int G)
{
  extern __shared__ float wt[];           // [32][cin] weight tile
  const int lane = threadIdx.x;
  const int half = lane >> 4;
  const int lm   = lane & 15;
  const int otiles = cout >> 5;
  const int lt = blockIdx.x / otiles;
  const int ot = blockIdx.x % otiles;
  const int b  = blockIdx.y;
  const int l0 = lt << 4;
  const int o0 = ot << 5;

  for (int i = lane; i < 32 * cin; i += 32)
    wt[i] = W[(long)(o0 + i / cin) * cin + (i % cin)];
  __syncthreads();

  const float gs0 = gam[o0 + lm] * BNS,      bs0 = bet[o0 + lm];
  const float gs1 = gam[o0 + 16 + lm] * BNS, bs1 = bet[o0 + 16 + lm];

  float mx0[8], mx1[8];
#pragma unroll
  for (int r = 0; r < 8; ++r) { mx0[r] = -3.4e38f; mx1[r] = -3.4e38f; }

  const long idx_base = (long)b * G * L + l0 + lm;
  const float* Xb = X + (long)b * xbs;
  const bool vec = (xcs == 1) && ((cin & 3) == 0);

  for (int g = 0; g < G; ++g) {
    const int row = idx[idx_base + (long)g * L];
    const float* xr = Xb + (long)row * xrs;
    v8f acc0 = {0.f, 0.f, 0.f, 0.f, 0.f, 0.f, 0.f, 0.f};
    v8f acc1 = {0.f, 0.f, 0.f, 0.f, 0.f, 0.f, 0.f, 0.f};
    if (vec) {                          // stages 2-4: contiguous channels
      for (int kb = 0; kb < cin; kb += 4) {
        const int ka = kb + 2 * half;
        const v2f av = *(const v2f*)(xr + ka);
        const v2f b0 = *(const v2f*)&wt[lm * cin + ka];
        const v2f b1 = *(const v2f*)&wt[(16 + lm) * cin + ka];
        acc0 = wmma4f32(av, b0, acc0);
        acc1 = wmma4f32(av, b1, acc1);
      }
    } else {                            // stage 1: cin=3, strided channels
      for (int kb = 0; kb < cin; kb += 4) {
        const int ka = kb + 2 * half;
        v2f av, b0, b1;
        av.x = (ka     < cin) ? xr[(long)ka * xcs]       : 0.f;
        av.y = (ka + 1 < cin) ? xr[(long)(ka + 1) * xcs] : 0.f;
        b0.x = (ka     < cin) ? wt[lm * cin + ka]        : 0.f;
        b0.y = (ka + 1 < cin) ? wt[lm * cin + ka + 1]    : 0.f;
        b1.x = (ka     < cin) ? wt[(16 + lm) * cin + ka]     : 0.f;
        b1.y = (ka + 1 < cin) ? wt[(16 + lm) * cin + ka + 1] : 0.f;
        acc0 = wmma4f32(av, b0, acc0);
        acc1 = wmma4f32(av, b1, acc1);
      }
    }
#pragma unroll
    for (int r = 0; r < 8; ++r) {
      float v0 = acc0[r] * gs0 + bs0;
      v0 = (v0 > 0.f) ? v0 : 0.2f * v0;
      mx0[r] = fmaxf(mx0[r], v0);
      float v1 = acc1[r] * gs1 + bs1;
      v1 = (v1 > 0.f) ? v1 : 0.2f * v1;
      mx1[r] = fmaxf(mx1[r], v1);
    }
  }
#pragma unroll
  for (int r = 0; r < 8; ++r) {
    const int mm = l0 + half * 8 + r;
    O[((long)b * L + mm) * cout + o0 + lm]      = mx0[r];
    O[((long)b * L + mm) * cout + o0 + 16 + lm] = mx1[r];
  }
}

// ---------------------------------------------------------------------------
// Row softmax in place over E[b, i, :n].  grid = (n, B), block = 128.
// ---------------------------------------------------------------------------
__global__ __launch_bounds__(128)
void softmax_rows(float* __restrict__ E, int n)
{
  __shared__ float red[128];
  const int tid = threadIdx.x;
  float* row = E + ((long)blockIdx.y * n + blockIdx.x) * n;

  float m = -3.4e38f;
  for (int j = tid; j < n; j += 128) m = fmaxf(m, row[j]);
  red[tid] = m; __syncthreads();
  for (int s = 64; s > 0; s >>= 1) { if (tid < s) red[tid] = fmaxf(red[tid], red[tid + s]); __syncthreads(); }
  m = red[0]; __syncthreads();

  float sum = 0.f;
  for (int j = tid; j < n; j += 128) { float e = __expf(row[j] - m); row[j] = e; sum += e; }
  red[tid] = sum; __syncthreads();
  for (int s = 64; s > 0; s >>= 1) { if (tid < s) red[tid] += red[tid + s]; __syncthreads(); }
  const float inv = 1.0f / red[0];
  for (int j = tid; j < n; j += 128) row[j] *= inv;
}

// rowscale[b,j] = 1 / (1e-9 + sum_i S[b,i,j])
__global__ void col_scale(const float* __restrict__ S, float* __restrict__ scl, int n)
{
  const int b = blockIdx.y;
  const int j = blockIdx.x * blockDim.x + threadIdx.x;
  if (j >= n) return;
  float s = 0.f;
  for (int i = 0; i < n; ++i) s += S[((long)b * n + i) * n + j];
  scl[(long)b * n + j] = 1.0f / (1e-9f + s);
}

// out[b, :c] = max_n H[b,n,c]; out[b, c:2c] = mean_n H[b,n,c]
__global__ void pool_maxmean(const float* __restrict__ H, float* __restrict__ out,
                             int n, int c)
{
  const int b  = blockIdx.y;
  const int ch = blockIdx.x * blockDim.x + threadIdx.x;
  if (ch >= c) return;
  float mx = -3.4e38f, sm = 0.f;
  for (int i = 0; i < n; ++i) {
    const float v = H[((long)b * n + i) * c + ch];
    mx = fmaxf(mx, v);
    sm += v;
  }
  out[(long)b * 2 * c + ch]     = mx;
  out[(long)b * 2 * c + c + ch] = sm / (float)n;
}

// ---------------------------------------------------------------------------
extern "C" void kernel_launch(void* const* d_in, const int* in_sizes, int n_in,
                              void* d_out, int out_size, void* d_ws, size_t ws_size,
                              hipStream_t stream)
{
  (void)in_sizes; (void)n_in; (void)out_size; (void)ws_size;
  const int NB = 16;

  const float* x    = (const float*)d_in[0];       // [16,3,2048]
  const int*   idxs[5] = {nullptr,
                          (const int*)d_in[1], (const int*)d_in[2],
                          (const int*)d_in[3], (const int*)d_in[4]};
  const float *w[6], *gm[6], *bt[6];
  for (int i = 1; i <= 5; ++i) {
    w[i]  = (const float*)d_in[5 + (i - 1) * 3 + 0];
    gm[i] = (const float*)d_in[5 + (i - 1) * 3 + 1];
    bt[i] = (const float*)d_in[5 + (i - 1) * 3 + 2];
  }
  struct TP { const float *qk, *vw, *vb, *tw, *tb, *g, *b; } tp[5];
  for (int i = 1; i <= 4; ++i) {
    const int base = 20 + (i - 1) * 7;
    tp[i] = { (const float*)d_in[base + 0], (const float*)d_in[base + 1],
              (const float*)d_in[base + 2], (const float*)d_in[base + 3],
              (const float*)d_in[base + 4], (const float*)d_in[base + 5],
              (const float*)d_in[base + 6] };
  }

  // workspace layout (floats)
  float* fA  = (float*)d_ws;                 // features ping  (<= 16*64*1024)
  float* fB  = fA  + (1l << 20);             // features pong
  float* Eb  = fB  + (1l << 20);             // attention E/S  (16*512*512)
  float* Qb  = Eb  + 16l * 512 * 512;        // q              (16*512*16 max)
  float* Vb  = Qb  + 16l * 512 * 16;         // v              (16*512*64 max)
  float* XMR = Vb  + 16l * 512 * 64;         // x - x_r
  float* SCL = XMR + 16l * 512 * 64;         // 1/(1e-9+colsum)

  auto gemm = [&](const float* A, long Abs, long As0, long As1,
                  const float* Bm, long Bbs, long Bs0, long Bs1,
                  float* O, long Obs, int M, int N, int K, int mode,
                  const float* bias, const float* gamv, const float* betv,
                  const float* resid, long Rbs, const float* rsc, long Sbs) {
    dim3 grid((unsigned)((M / 32) * (N / 16)), (unsigned)NB);
    wmma_gemm<<<grid, dim3(32), 0, stream>>>(A, Abs, As0, As1, Bm, Bbs, Bs0, Bs1,
                                             O, Obs, M, N, K, mode,
                                             bias, gamv, betv, resid, Rbs, rsc, Sbs);
  };

  auto run_trans = [&](const float* xin, int n, int c, const TP& t, float* xout) {
    const int q4 = c / 4;
    // q = x @ qk^T                                  [n, q4]
    gemm(xin, (long)n * c, c, 1, t.qk, 0, c, 1,
         Qb, (long)n * q4, n, q4, c, 0, nullptr, nullptr, nullptr, nullptr, 0, nullptr, 0);
    // E = q @ q^T                                   [n, n]
    gemm(Qb, (long)n * q4, q4, 1, Qb, (long)n * q4, q4, 1,
         Eb, (long)n * n, n, n, q4, 0, nullptr, nullptr, nullptr, nullptr, 0, nullptr, 0);
    softmax_rows<<<dim3(n, NB), 128, 0, stream>>>(Eb, n);
    col_scale<<<dim3((n + 127) / 128, NB), 128, 0, stream>>>(Eb, SCL, n);
    // v = x @ vw^T + vb                             [n, c]
    gemm(xin, (long)n * c, c, 1, t.vw, 0, c, 1,
         Vb, (long)n * c, n, c, c, 1, t.vb, nullptr, nullptr, nullptr, 0, nullptr, 0);
    // xmr[j,cc] = x[j,cc] - rowscale[j]*sum_i S[i,j] v[i,cc]   (A=S^T, B=v^T)
    gemm(Eb, (long)n * n, 1, n, Vb, (long)n * c, 1, c,
         XMR, (long)n * c, n, c, n, 2, nullptr, nullptr, nullptr,
         xin, (long)n * c, SCL, n);
    // out = x + relu(bn(xmr @ tw^T + tb))
    gemm(XMR, (long)n * c, c, 1, t.tw, 0, c, 1,
         xout, (long)n * c, n, c, c, 3, t.tb, t.g, t.b, xin, (long)n * c, nullptr, 0);
  };

  // stage1: x raw [16,3,2048]: elem(b,row,ch) = x[b*6144 + ch*2048 + row]
  stage_gcmax<<<dim3((512 / 16) * (64 / 32), NB), 32, 32 * 3 * 4, stream>>>(
      x, 6144, 1, 2048, idxs[1], w[1], gm[1], bt[1], fA, 512, 3, 64, 64);
  run_trans(fA, 512, 64, tp[1], fB);

  stage_gcmax<<<dim3((256 / 16) * (128 / 32), NB), 32, 32 * 64 * 4, stream>>>(
      fB, 512l * 64, 64, 1, idxs[2], w[2], gm[2], bt[2], fA, 256, 64, 128, 64);
  run_trans(fA, 256, 128, tp[2], fB);

  stage_gcmax<<<dim3((128 / 16) * (256 / 32), NB), 32, 32 * 128 * 4, stream>>>(
      fB, 256l * 128, 128, 1, idxs[3], w[3], gm[3], bt[3], fA, 128, 128, 256, 64);
  run_trans(fA, 128, 256, tp[3], fB);

  stage_gcmax<<<dim3((64 / 16) * (512 / 32), NB), 32, 32 * 256 * 4, stream>>>(
      fB, 128l * 256, 256, 1, idxs[4], w[4], gm[4], bt[4], fA, 64, 256, 512, 64);
  run_trans(fA, 64, 512, tp[4], fB);

  // conv5 + bn + leaky relu:  fB [16,64,512] @ w5[1024,512]^T -> fA [16,64,1024]
  gemm(fB, 64l * 512, 512, 1, w[5], 0, 512, 1,
       fA, 64l * 1024, 64, 1024, 512, 4, nullptr, gm[5], bt[5], nullptr, 0, nullptr, 0);

  pool_maxmean<<<dim3(1024 / 256, NB), 256, 0, stream>>>(fA, (float*)d_out, 64, 1024);
}